// GCN_35381940584979
// MI455X (gfx1250) — compile-verified
//
#include <hip/hip_runtime.h>
#include <stdint.h>

#define NNODES 50000
#define NEDGES 800000
#define IN_DIM 256
#define HID    128
#define OUT_DIM 64
#define NLAYER 3
#define MAXKK  32

typedef __attribute__((ext_vector_type(2))) float v2f;
typedef __attribute__((ext_vector_type(8))) float v8f;

// ---------------------------------------------------------------- utilities
__global__ void k_zero(float* __restrict__ p, int n) {
    int i = blockIdx.x * blockDim.x + threadIdx.x;
    if (i < n) p[i] = 0.0f;
}

__global__ void k_degree(const int* __restrict__ src, const int* __restrict__ dst,
                         float* __restrict__ dsrc, float* __restrict__ ddst, int E) {
    int e = blockIdx.x * blockDim.x + threadIdx.x;
    if (e < E) {
        atomicAdd(&dsrc[src[e]], 1.0f);
        atomicAdd(&ddst[dst[e]], 1.0f);
    }
}

__global__ void k_rsqrt(float* __restrict__ p, int n) {
    int i = blockIdx.x * blockDim.x + threadIdx.x;
    if (i < n) p[i] = rsqrtf(fmaxf(p[i], 1.0f));
}

// ------------------------------------------------ fp32 WMMA GEMM: Out = A @ W^T + bias
// A: [M, Kdim] row-major (lda), W: [ncols, Kdim] row-major (ldw), Out: [M, ncols] (ldo).
// One wave computes a 16x16 tile via v_wmma_f32_16x16x4_f32.
// A-frag (16x4 f32): lane l -> M = l&15, K = 2*(l>>4) + v   (v = vgpr 0/1)
// B-frag (4x16 f32): lane l -> N = l&15, K = 2*(l>>4) + v ; B[k][n] = W[n][k]
// C/D  (16x16 f32): lane l, vgpr v -> M = v + 8*(l>>4), N = l&15
__global__ __launch_bounds__(256) void k_gemm_wmma(
    const float* __restrict__ A, int lda,
    const float* __restrict__ W, int ldw,
    const float* __restrict__ bias,
    float* __restrict__ Out, int ldo,
    int Mtiles, int Ntiles, int Kdim, int relu)
{
    int wave = threadIdx.x >> 5;
    int lane = threadIdx.x & 31;
    int t = blockIdx.x * 8 + wave;
    if (t >= Mtiles * Ntiles) return;                 // uniform per wave
    int nt = t % Ntiles;
    int mt = t / Ntiles;

    int lrow  = lane & 15;
    int khalf = (lane >> 4) << 1;                     // 0 or 2

    const float* ap = A + (size_t)(mt * 16 + lrow) * lda + khalf;
    const float* bp = W + (size_t)(nt * 16 + lrow) * ldw + khalf;

    v8f c0 = {};
    v8f c1 = {};
    int k0 = 0;
    #pragma unroll 4
    for (; k0 + 8 <= Kdim; k0 += 8) {
        v2f a0 = *(const v2f*)(ap + k0);
        v2f b0 = *(const v2f*)(bp + k0);
        v2f a1 = *(const v2f*)(ap + k0 + 4);
        v2f b1 = *(const v2f*)(bp + k0 + 4);
        c0 = __builtin_amdgcn_wmma_f32_16x16x4_f32(false, a0, false, b0, (short)0, c0, false, false);
        c1 = __builtin_amdgcn_wmma_f32_16x16x4_f32(false, a1, false, b1, (short)0, c1, false, false);
    }
    for (; k0 < Kdim; k0 += 4) {
        v2f a0 = *(const v2f*)(ap + k0);
        v2f b0 = *(const v2f*)(bp + k0);
        c0 = __builtin_amdgcn_wmma_f32_16x16x4_f32(false, a0, false, b0, (short)0, c0, false, false);
    }

    int col = nt * 16 + lrow;
    float bv = bias[col];
    int rbase = mt * 16 + ((lane >> 4) ? 8 : 0);
    #pragma unroll
    for (int v = 0; v < 8; ++v) {
        float o = c0[v] + c1[v] + bv;
        if (relu) o = fmaxf(o, 0.0f);
        Out[(size_t)(rbase + v) * ldo + col] = o;
    }
}

// ------------------------------------------------ MaxK: top-32 of 128 per row,
// scale survivors by norm_src, compact to (cval[row][32], cidx[row][32]).
__device__ __forceinline__ unsigned fkey(float f) {
    unsigned u = __float_as_uint(f);
    return (u & 0x80000000u) ? ~u : (u | 0x80000000u);   // order-preserving map
}

__global__ __launch_bounds__(256) void k_maxk(
    const float* __restrict__ G, const float* __restrict__ nsrc,
    float* __restrict__ cval, unsigned char* __restrict__ cidx, int n)
{
    int wave = threadIdx.x >> 5;
    int lane = threadIdx.x & 31;
    int row = blockIdx.x * 8 + wave;
    if (row >= n) return;

    const float4 v4 = *(const float4*)(G + (size_t)row * HID + lane * 4);
    float vv[4] = {v4.x, v4.y, v4.z, v4.w};
    unsigned kk[4];
    #pragma unroll
    for (int j = 0; j < 4; ++j) kk[j] = fkey(vv[j]);

    // bitwise radix-select: thr = 32nd-largest key (max t with count(key>=t) >= 32)
    unsigned thr = 0u;
    for (int bit = 31; bit >= 0; --bit) {
        unsigned cand = thr | (1u << bit);
        int cnt = 0;
        #pragma unroll
        for (int j = 0; j < 4; ++j) cnt += (kk[j] >= cand) ? 1 : 0;
        #pragma unroll
        for (int off = 16; off > 0; off >>= 1) cnt += __shfl_xor(cnt, off, 32);
        if (cnt >= MAXKK) thr = cand;
    }

    // compaction: exclusive scan of per-lane keep counts via shuffles
    int keepj[4];
    int mykeep = 0;
    #pragma unroll
    for (int j = 0; j < 4; ++j) { keepj[j] = (kk[j] >= thr) ? 1 : 0; mykeep += keepj[j]; }
    int incl = mykeep;
    #pragma unroll
    for (int off = 1; off < 32; off <<= 1) {
        int t = __shfl_up(incl, off, 32);
        if (lane >= off) incl += t;
    }
    int p = incl - mykeep;                             // exclusive prefix
    float ns = nsrc[row];
    #pragma unroll
    for (int j = 0; j < 4; ++j) {
        if (keepj[j]) {
            if (p < MAXKK) {
                cval[(size_t)row * MAXKK + p] = vv[j] * ns;
                cidx[(size_t)row * MAXKK + p] = (unsigned char)(lane * 4 + j);
            }
            ++p;
        }
    }
}

// ------------------------------------------------ sparse scatter: 1 wave / edge,
// 32 lanes = the 32 surviving channels -> atomic add into acc[dst]
__global__ __launch_bounds__(256) void k_scatter(
    const float* __restrict__ cval, const unsigned char* __restrict__ cidx,
    const int* __restrict__ src, const int* __restrict__ dst,
    float* __restrict__ acc, int E)
{
    int g = blockIdx.x * blockDim.x + threadIdx.x;
    int e = g >> 5;
    int lane = g & 31;
    if (e >= E) return;                                // uniform per wave
    int s = src[e];
    int d = dst[e];
    float v = cval[(size_t)s * MAXKK + lane];
    int   c = cidx[(size_t)s * MAXKK + lane];
    atomicAdd(acc + (size_t)d * HID + c, v);
}

// ------------------------------------------------ h = acc * norm_dst + gcn_b
__global__ void k_finalize(const float* __restrict__ acc, const float* __restrict__ ndst,
                           const float* __restrict__ gb, float* __restrict__ h, int total)
{
    int i = blockIdx.x * blockDim.x + threadIdx.x;
    if (i < total) {
        int node = i >> 7;      // /HID
        int c    = i & 127;
        h[i] = acc[i] * ndst[node] + gb[c];
    }
}

// ---------------------------------------------------------------- launcher
extern "C" void kernel_launch(void* const* d_in, const int* in_sizes, int n_in,
                              void* d_out, int out_size, void* d_ws, size_t ws_size,
                              hipStream_t stream) {
    const float* x         = (const float*)d_in[0];
    const int*   src       = (const int*)  d_in[1];
    const int*   dst       = (const int*)  d_in[2];
    const float* lin_in_w  = (const float*)d_in[3];
    const float* lin_in_b  = (const float*)d_in[4];
    const float* lin_w     = (const float*)d_in[5];
    const float* lin_b     = (const float*)d_in[6];
    const float* gcn_b     = (const float*)d_in[7];
    const float* lin_out_w = (const float*)d_in[8];
    const float* lin_out_b = (const float*)d_in[9];
    float* out = (float*)d_out;

    float* ws        = (float*)d_ws;
    float* norm_src  = ws;                                  // N
    float* norm_dst  = norm_src + NNODES;                   // N
    float* h         = norm_dst + NNODES;                   // N*HID
    float* acc       = h   + (size_t)NNODES * HID;          // N*HID
    float* cval      = acc + (size_t)NNODES * HID;          // N*32
    unsigned char* cidx = (unsigned char*)(cval + (size_t)NNODES * MAXKK); // N*32 bytes

    const int nh = NNODES * HID;

    // degrees -> norms
    k_zero<<<(2 * NNODES + 255) / 256, 256, 0, stream>>>(norm_src, 2 * NNODES);
    k_degree<<<(NEDGES + 255) / 256, 256, 0, stream>>>(src, dst, norm_src, norm_dst, NEDGES);
    k_rsqrt<<<(2 * NNODES + 255) / 256, 256, 0, stream>>>(norm_src, 2 * NNODES);

    // h = relu(x @ lin_in_w^T + b)   M=50000 -> 3125 M-tiles, 8 N-tiles
    k_gemm_wmma<<<3125, 256, 0, stream>>>(x, IN_DIM, lin_in_w, IN_DIM, lin_in_b,
                                          h, HID, 3125, HID / 16, IN_DIM, 1);

    for (int i = 0; i < NLAYER; ++i) {
        // g = h @ lin_w[i]^T + lin_b[i]  -> acc (temporary)
        k_gemm_wmma<<<3125, 256, 0, stream>>>(h, HID, lin_w + (size_t)i * HID * HID, HID,
                                              lin_b + (size_t)i * HID,
                                              acc, HID, 3125, HID / 16, HID, 0);
        // maxk + fold norm_src, compact to 32 channels
        k_maxk<<<(NNODES + 7) / 8, 256, 0, stream>>>(acc, norm_src, cval, cidx, NNODES);
        // acc <- 0, then sparse scatter-add
        k_zero<<<(nh + 255) / 256, 256, 0, stream>>>(acc, nh);
        k_scatter<<<((size_t)NEDGES * 32 + 255) / 256, 256, 0, stream>>>(cval, cidx, src, dst, acc, NEDGES);
        // h = acc * norm_dst + gcn_b[i]
        k_finalize<<<(nh + 255) / 256, 256, 0, stream>>>(acc, norm_dst, gcn_b + (size_t)i * HID, h, nh);
    }

    // out = h @ lin_out_w^T + b   (4 N-tiles)
    k_gemm_wmma<<<(3125 * 4 + 7) / 8, 256, 0, stream>>>(h, HID, lin_out_w, HID, lin_out_b,
                                                        out, OUT_DIM, 3125, OUT_DIM / 16, HID, 0);
}